// ConvMessageAggregator_60318520705542
// MI455X (gfx1250) — compile-verified
//
#include <hip/hip_runtime.h>

// ConvMessageAggregator on MI455X (gfx1250).
// Bandwidth-bound streaming kernel: async global->LDS DMA (CDNA5 ASYNCcnt path)
// with non-temporal load hint (single-use 451 MB stream, don't churn L2),
// double-buffered over n, non-temporal stores. No WMMA: AI ~1.6 flop/byte and
// the per-element ReLU chain cannot be expressed as a GEMM; HBM (23.3 TB/s)
// is the only limiter (~21 us floor).

#define MSGW 688
#define LSEQ 16
#define ROWS 10                        // x rows l = 6..15
#define TILE_BYTES (ROWS * MSGW * 4)   // 27,520 contiguous bytes per n
#define NPB 8                          // n per block
#define THREADS 256
#define XFERS (TILE_BYTES / 16)        // 1720 b128 lane-transfers per tile
#define CHUNKS ((XFERS + THREADS - 1) / THREADS) // 7

__device__ __forceinline__ void async_copy_tile(const float* gbase,
                                                uint32_t lds_base, int tid) {
#pragma unroll
  for (int c = 0; c < CHUNKS; ++c) {
    int i = c * THREADS + tid;
    if (i < XFERS) {
      uint32_t lds_addr = lds_base + (uint32_t)(i * 16);
      const char* g = (const char*)gbase + (size_t)i * 16;
      // VDST = per-lane LDS byte address, v[addr:addr+1] = 64-bit global
      // address. TH_LOAD_NT: single-use stream, bypass L2 retention.
      asm volatile("global_load_async_to_lds_b128 %0, %1, off th:TH_LOAD_NT"
                   :: "v"(lds_addr), "v"(g)
                   : "memory");
    }
  }
}

__global__ __launch_bounds__(THREADS)
void ConvMessageAggregator_60318520705542_kernel(
    const float* __restrict__ msgs,
    const float* __restrict__ c1w, const float* __restrict__ c1b,
    const float* __restrict__ c2w, const float* __restrict__ c2b,
    const float* __restrict__ mw,  const float* __restrict__ mb,
    float* __restrict__ out, int N) {
  __shared__ alignas(16) float smem[2 * ROWS * MSGW];  // 55,040 B, double buffer

  const int tid = threadIdx.x;
  const int n0  = blockIdx.x * NPB;

  // low 32 bits of the generic pointer = wave-relative LDS byte offset
  const uint32_t lds0 = (uint32_t)(uintptr_t)(&smem[0]);
  const uint32_t lds1 = (uint32_t)(uintptr_t)(&smem[ROWS * MSGW]);

  // uniform weights (served by scalar/L2 cache)
  const float w10 = c1w[0], w11 = c1w[1], b1 = c1b[0];
  const float w20 = c2w[0], w21 = c2w[1], b2 = c2b[0];
  float wk[6];
#pragma unroll
  for (int k = 0; k < 6; ++k) wk[k] = mw[k];
  const float b3 = mb[0];

  // prologue: stage tile for n0 into buffer 0
  if (n0 < N)
    async_copy_tile(msgs + ((size_t)n0 * LSEQ + 6) * MSGW, lds0, tid);

  for (int i = 0; i < NPB; ++i) {
    const int n = n0 + i;
    if (n >= N) break;  // uniform within block

    // wait for this wave's async copies of the current buffer, then sync block
    asm volatile("s_wait_asynccnt 0x0" ::: "memory");
    __syncthreads();

    // prefetch next n into the alternate buffer (fully consumed at iter i-1)
    const int nn = n + 1;
    if (i + 1 < NPB && nn < N)
      async_copy_tile(msgs + ((size_t)nn * LSEQ + 6) * MSGW,
                      (i & 1) ? lds0 : lds1, tid);

    const float* buf = &smem[(i & 1) ? (ROWS * MSGW) : 0];

    for (int m = tid; m < MSGW; m += THREADS) {
      float x[10];  // x[l] = messages[n, l+6, m]
#pragma unroll
      for (int l = 0; l < 10; ++l) x[l] = buf[l * MSGW + m];

      float f1[8];  // f1[t] <-> conv1 output t+6, t=0..7
#pragma unroll
      for (int t = 0; t < 8; ++t)
        f1[t] = fmaxf(fmaf(w11, x[t + 2], fmaf(w10, x[t], b1)), 0.0f);

      float f2[6];  // f2[t] <-> conv2 output t+6 (= feat[k=t])
#pragma unroll
      for (int t = 0; t < 6; ++t)
        f2[t] = fmaxf(fmaf(w21, f1[t + 2], fmaf(w20, f1[t], b2)), 0.0f);

      float acc = b3;
#pragma unroll
      for (int k = 0; k < 6; ++k) acc = fmaf(wk[k], f2[k], acc);

      __builtin_nontemporal_store(fmaxf(acc, 0.0f),
                                  &out[(size_t)n * MSGW + m]);
    }
    // all waves done reading `buf` before it becomes a prefetch target again
    __syncthreads();
  }
}

extern "C" void kernel_launch(void* const* d_in, const int* in_sizes, int n_in,
                              void* d_out, int out_size, void* d_ws, size_t ws_size,
                              hipStream_t stream) {
  const float* msgs = (const float*)d_in[0];
  const float* c1w  = (const float*)d_in[1];
  const float* c1b  = (const float*)d_in[2];
  const float* c2w  = (const float*)d_in[3];
  const float* c2b  = (const float*)d_in[4];
  const float* mw   = (const float*)d_in[5];
  const float* mb   = (const float*)d_in[6];
  float* out = (float*)d_out;

  const int N = in_sizes[0] / (LSEQ * MSGW);  // 16384
  const int nblocks = (N + NPB - 1) / NPB;    // 2048

  ConvMessageAggregator_60318520705542_kernel<<<nblocks, THREADS, 0, stream>>>(
      msgs, c1w, c1b, c2w, c2b, mw, mb, out, N);
}